// gridded_nufft_48704929136777
// MI455X (gfx1250) — compile-verified
//
#include <hip/hip_runtime.h>
#include <hip/hip_bf16.h>

// ---------------------------------------------------------------------------
// Gridded NUFFT forward for MI455X (gfx1250, wave32, WMMA).
//   1) Build centered-DFT matrix A (320x256 complex) folding pad+fftshifts.
//   2) Per coil: T1 = A @ img   (complex GEMM via V_WMMA_F32_16X16X4_F32)
//   3) Per coil: G  = T1 @ A^T  (complex GEMM via V_WMMA_F32_16X16X4_F32)
//   4) Nearest-neighbor gather of G at rounded trajectory points.
// fp32 WMMA keeps FFT-grade precision; the whole pipeline is bandwidth-bound
// (~130 MB moved -> ~6us floor at 23.3 TB/s), compute is ~3 GFLOP.
// ---------------------------------------------------------------------------

typedef float v2f __attribute__((ext_vector_type(2)));
typedef float v8f __attribute__((ext_vector_type(8)));

#define HOS 320      // oversampled grid size (both dims)
#define HIN 256      // input image size (both dims)
#define NCOIL 8
#define NPTS (64 * 16384)

// A[k][n] = exp(-2*pi*i * (k-160)*(n-128) / 320) / sqrt(320)
// (k in [0,320): output freq after fftshift; n in [0,256): unpadded spatial,
//  pad offset 32 and ifftshift folded in: (n+32)-160 = n-128)
__global__ void nufft_dft_build(float* __restrict__ A) {
    int i = blockIdx.x * blockDim.x + threadIdx.x;
    if (i >= HOS * HIN) return;
    int k = i / HIN;
    int n = i % HIN;
    int m = (k - 160) * (n - 128);
    int r = m % HOS;                 // exact integer phase reduction
    if (r < 0) r += HOS;
    const float twopi_over = 6.283185307179586f / (float)HOS;
    float th = -twopi_over * (float)r;
    float s, c;
    sincosf(th, &s, &c);
    const float scale = 0.05590169943749474f;  // 1/sqrt(320)
    A[2 * i + 0] = c * scale;
    A[2 * i + 1] = s * scale;
}

// ---- complex GEMM tile: D(16x16) += A(16x4) x B(4x16) over K, 4 wmma/step --
// A-operand layout (f32 16x4): lane L: M = L&15, VGPR r holds K = 2*(L>>4)+r
// B-operand layout (f32 4x16): lane L: N = L&15, VGPR r holds K = 2*(L>>4)+r
// C/D layout: lane L: N = L&15, VGPR v holds M = v + 8*(L>>4)

// T1[c] (320x256) = A (320x256) @ imgC[c] (256x256); img given as re/im planes
__global__ void nufft_gemm1(const float* __restrict__ Adft,
                            const float* __restrict__ imr,
                            const float* __restrict__ imi,
                            float* __restrict__ T1) {
    const int M = HOS, K = HIN, N = HIN;
    int lane = threadIdx.x;
    int row  = lane & 15;
    int grp  = lane >> 4;
    int tm   = blockIdx.x * 16;
    int tn   = blockIdx.y * 16;
    int c    = blockIdx.z;

    const float* br_ = imr + (size_t)c * K * N;
    const float* bi_ = imi + (size_t)c * K * N;

    v8f RR = {}, II = {}, RI = {}, IR = {};
    for (int k0 = 0; k0 < K; k0 += 4) {
        int ka = k0 + 2 * grp;
        const float* ap = Adft + 2 * ((size_t)(tm + row) * K + ka);
        v2f ar = { ap[0], ap[2] };
        v2f ai = { ap[1], ap[3] };
        const float* bpr = br_ + (size_t)ka * N + tn + row;
        const float* bpi = bi_ + (size_t)ka * N + tn + row;
        v2f br = { bpr[0], bpr[N] };
        v2f bi = { bpi[0], bpi[N] };
        RR = __builtin_amdgcn_wmma_f32_16x16x4_f32(false, ar, false, br, (short)0, RR, false, false);
        II = __builtin_amdgcn_wmma_f32_16x16x4_f32(false, ai, false, bi, (short)0, II, false, false);
        RI = __builtin_amdgcn_wmma_f32_16x16x4_f32(false, ar, false, bi, (short)0, RI, false, false);
        IR = __builtin_amdgcn_wmma_f32_16x16x4_f32(false, ai, false, br, (short)0, IR, false, false);
    }
    for (int v = 0; v < 8; ++v) {
        int m = tm + v + 8 * grp;
        int n = tn + row;
        float* o = T1 + 2 * (((size_t)c * M + m) * N + n);
        o[0] = RR[v] - II[v];
        o[1] = RI[v] + IR[v];
    }
}

// G[c] (320x320) = T1[c] (320x256) @ A^T (256x320);  B[x][kx] = Adft[kx*256+x]
__global__ void nufft_gemm2(const float* __restrict__ T1,
                            const float* __restrict__ Adft,
                            float* __restrict__ G) {
    const int M = HOS, K = HIN, N = HOS;
    int lane = threadIdx.x;
    int row  = lane & 15;
    int grp  = lane >> 4;
    int tm   = blockIdx.x * 16;
    int tn   = blockIdx.y * 16;
    int c    = blockIdx.z;

    v8f RR = {}, II = {}, RI = {}, IR = {};
    for (int k0 = 0; k0 < K; k0 += 4) {
        int ka = k0 + 2 * grp;
        const float* ap = T1 + 2 * (((size_t)c * M + tm + row) * K + ka);
        v2f ar = { ap[0], ap[2] };
        v2f ai = { ap[1], ap[3] };
        // B element (k=ka+r, n=tn+row) = Adft[(tn+row)*K + ka + r] : contiguous
        const float* bp = Adft + 2 * ((size_t)(tn + row) * K + ka);
        v2f br = { bp[0], bp[2] };
        v2f bi = { bp[1], bp[3] };
        RR = __builtin_amdgcn_wmma_f32_16x16x4_f32(false, ar, false, br, (short)0, RR, false, false);
        II = __builtin_amdgcn_wmma_f32_16x16x4_f32(false, ai, false, bi, (short)0, II, false, false);
        RI = __builtin_amdgcn_wmma_f32_16x16x4_f32(false, ar, false, bi, (short)0, RI, false, false);
        IR = __builtin_amdgcn_wmma_f32_16x16x4_f32(false, ai, false, br, (short)0, IR, false, false);
    }
    for (int v = 0; v < 8; ++v) {
        int m = tm + v + 8 * grp;
        int n = tn + row;
        float* o = G + 2 * (((size_t)c * M + m) * N + n);
        o[0] = RR[v] - II[v];
        o[1] = RI[v] + IR[v];
    }
}

// out[c][p] = G[c][round(clip(t0*1.25+160,0,319))][round(clip(t1*1.25+160,0,319))]
__global__ void nufft_gather(const float* __restrict__ trj,
                             const float* __restrict__ G,
                             float* __restrict__ out) {
    int p = blockIdx.x * blockDim.x + threadIdx.x;
    if (p >= NPTS) return;
    float t0 = trj[2 * p + 0];
    float t1 = trj[2 * p + 1];
    float r0 = fminf(fmaxf(t0 * 1.25f + 160.0f, 0.0f), 319.0f);
    float r1 = fminf(fmaxf(t1 * 1.25f + 160.0f, 0.0f), 319.0f);
    int i0 = (int)rintf(r0);   // round-half-even, matches jnp.round
    int i1 = (int)rintf(r1);
    int flat = i0 * HOS + i1;
#pragma unroll
    for (int c = 0; c < NCOIL; ++c) {
        const float* g = G + 2 * ((size_t)c * HOS * HOS + flat);
        float* o = out + 2 * ((size_t)c * NPTS + p);
        o[0] = g[0];
        o[1] = g[1];
    }
}

extern "C" void kernel_launch(void* const* d_in, const int* in_sizes, int n_in,
                              void* d_out, int out_size, void* d_ws, size_t ws_size,
                              hipStream_t stream) {
    (void)in_sizes; (void)n_in; (void)out_size; (void)ws_size;
    const float* imr = (const float*)d_in[0];   // (1,8,256,256) f32
    const float* imi = (const float*)d_in[1];   // (1,8,256,256) f32
    const float* trj = (const float*)d_in[2];   // (1,64,16384,2) f32
    float* out = (float*)d_out;                  // (1,8,64,16384) complex64 -> f32 pairs

    float* ws = (float*)d_ws;
    float* A  = ws;                                  // 320*256*2      = 163,840 f
    float* T1 = ws + 163840;                         // 8*320*256*2    = 1,310,720 f
    float* G  = ws + 163840 + 1310720;               // 8*320*320*2    = 1,638,400 f
    // total ws use: 3,112,960 floats = ~12.5 MB

    nufft_dft_build<<<(HOS * HIN + 255) / 256, 256, 0, stream>>>(A);

    dim3 g1(HOS / 16, HIN / 16, NCOIL);   // 20 x 16 x 8 tiles
    nufft_gemm1<<<g1, 32, 0, stream>>>(A, imr, imi, T1);

    dim3 g2(HOS / 16, HOS / 16, NCOIL);   // 20 x 20 x 8 tiles
    nufft_gemm2<<<g2, 32, 0, stream>>>(T1, A, G);

    nufft_gather<<<(NPTS + 255) / 256, 256, 0, stream>>>(trj, G, out);
}